// SimpleMHA2D_60404420050968
// MI455X (gfx1250) — compile-verified
//
#include <hip/hip_runtime.h>
#include <math.h>

typedef __attribute__((ext_vector_type(2))) float v2f;
typedef __attribute__((ext_vector_type(8))) float v8f;

#define S_TOT 16384   // H*W
#define C_DIM 256
#define B_DIM 16

// ---------------------------------------------------------------------------
// Kernel 0: fold the learned query into Wk / bk.
//   wq[c][h]  = sum_d Wk[c, h*64+d] * q[h,d]   (heads padded to 16 with zeros)
//   bq[h]     = sum_d bk[h*64+d]    * q[h,d]
// ---------------------------------------------------------------------------
__global__ __launch_bounds__(256) void prep_kernel(const float* __restrict__ Wk,
                                                   const float* __restrict__ bk,
                                                   const float* __restrict__ q,
                                                   float* __restrict__ wq,
                                                   float* __restrict__ bq) {
    int c = threadIdx.x;  // 256 threads, one channel each
    for (int h = 0; h < 8; ++h) {
        const float* wrow = Wk + (size_t)c * 512 + h * 64;
        const float* qrow = q + h * 64;
        float s = 0.f;
        for (int d = 0; d < 64; ++d) s += wrow[d] * qrow[d];
        wq[c * 16 + h]     = s;
        wq[c * 16 + 8 + h] = 0.f;  // pad heads 8..15
    }
    if (c < 16) {
        float s = 0.f;
        if (c < 8) {
            for (int d = 0; d < 64; ++d) s += bk[c * 64 + d] * q[c * 64 + d];
        }
        bq[c] = s;
    }
}

// ---------------------------------------------------------------------------
// Kernel 1: scores[b,s,h] = kv[b,s,:] @ wq[:,h] + bq[h]
// WMMA f32 16x16x4: A = 16 s-rows x 4 channels, B = 4 channels x 16 heads.
// One wave per 16 spatial rows; 64 WMMAs accumulate over C=256.
// ---------------------------------------------------------------------------
__global__ __launch_bounds__(256) void scores_kernel(const float* __restrict__ kv,
                                                     const float* __restrict__ wq,
                                                     const float* __restrict__ bq,
                                                     float* __restrict__ scores) {
    __shared__ float lwq[4112];  // 4096 wq + 16 bq
    int tid = threadIdx.x;
    for (int i = tid; i < 4112; i += 256)
        lwq[i] = (i < 4096) ? wq[i] : bq[i - 4096];
    __syncthreads();

    int b    = blockIdx.y;
    int wave = tid >> 5;
    int lane = tid & 31;
    int half = lane >> 4;    // K split across lane halves (16x4 f32 A layout)
    int lm   = lane & 15;
    int s0   = blockIdx.x * 128 + wave * 16;

    const float* kvrow = kv + ((size_t)b * S_TOT + s0 + lm) * C_DIM;

    v8f acc = {};
    for (int kc = 0; kc < 64; ++kc) {
        int cb = kc * 4 + half * 2;
        float2 av = *(const float2*)(kvrow + cb);           // A: M=lm, K=cb..cb+1
        v2f a;  a.x = av.x;            a.y = av.y;
        v2f bm; bm.x = lwq[cb * 16 + lm];                    // B: K=cb,   N=lm
                bm.y = lwq[(cb + 1) * 16 + lm];              // B: K=cb+1, N=lm
        acc = __builtin_amdgcn_wmma_f32_16x16x4_f32(
            false, a, false, bm, (short)0, acc, false, false);
    }

    if (lm < 8) {  // only real heads stored; stride-8 score rows
        float bqv = lwq[4096 + lm];
        for (int r = 0; r < 8; ++r) {
            int M = r + half * 8;  // C/D layout: VGPR r, lanes 0-15 -> M=r, 16-31 -> M=r+8
            scores[((size_t)b * S_TOT + s0 + M) * 8 + lm] = acc[r] + bqv;
        }
    }
}

// ---------------------------------------------------------------------------
// Kernel 2: per-(b,h) softmax stats over S: max and sum(exp).
// ---------------------------------------------------------------------------
__global__ __launch_bounds__(256) void stats_kernel(const float* __restrict__ scores,
                                                    float* __restrict__ maxv,
                                                    float* __restrict__ denom) {
    int b = blockIdx.x >> 3;
    int h = blockIdx.x & 7;
    __shared__ float red[256];
    int t = threadIdx.x;
    const size_t base = (size_t)b * S_TOT * 8 + h;

    float m = -1e30f;
    for (int s = t; s < S_TOT; s += 256)
        m = fmaxf(m, scores[base + (size_t)s * 8]);
    red[t] = m;
    __syncthreads();
    for (int off = 128; off > 0; off >>= 1) {
        if (t < off) red[t] = fmaxf(red[t], red[t + off]);
        __syncthreads();
    }
    float gm = red[0];
    __syncthreads();

    float sum = 0.f;
    for (int s = t; s < S_TOT; s += 256)
        sum += __expf(scores[base + (size_t)s * 8] - gm);
    red[t] = sum;
    __syncthreads();
    for (int off = 128; off > 0; off >>= 1) {
        if (t < off) red[t] += red[t + off];
        __syncthreads();
    }
    if (t == 0) {
        maxv[b * 16 + h]  = gm;
        denom[b * 16 + h] = red[0];
    }
}

// ---------------------------------------------------------------------------
// Kernel 3: pkv[b,h,c] = sum_s softmax(scores)[b,s,h] * kv[b,s,c]
// WMMA f32 16x16x4: A = 16 heads (8 padded) x 4 spatial probs, B = 4 s x 16 c.
// S split into 64 chunks of 256; partial sums combined with f32 atomics.
// ---------------------------------------------------------------------------
__global__ __launch_bounds__(256) void pkv_kernel(const float* __restrict__ kv,
                                                  const float* __restrict__ scores,
                                                  const float* __restrict__ maxv,
                                                  const float* __restrict__ denom,
                                                  float* __restrict__ pkv) {
    int tid  = threadIdx.x;
    int wave = tid >> 5;
    int lane = tid & 31;
    int half = lane >> 4;
    int lm   = lane & 15;

    int c0 = blockIdx.x * 16;
    int bb = blockIdx.z;
    int s_begin = (blockIdx.y * 8 + wave) * 256;

    float m    = (lm < 8) ? maxv[bb * 16 + lm] : 0.f;
    float invd = (lm < 8) ? 1.f / denom[bb * 16 + lm] : 0.f;  // zero kills pad heads

    const size_t kvbase = (size_t)bb * S_TOT * C_DIM;
    const size_t scbase = (size_t)bb * S_TOT * 8;

    v8f acc = {};
    for (int ks = 0; ks < 64; ++ks) {
        int sA = s_begin + ks * 4 + half * 2;
        float sc0 = (lm < 8) ? scores[scbase + (size_t)sA * 8 + lm] : 0.f;
        float sc1 = (lm < 8) ? scores[scbase + (size_t)(sA + 1) * 8 + lm] : 0.f;
        v2f a;  a.x = __expf(sc0 - m) * invd;   // A: M=head lm, K=spatial
                a.y = __expf(sc1 - m) * invd;
        const float* kvs = kv + kvbase + (size_t)sA * C_DIM + c0;
        v2f bm; bm.x = kvs[lm];                 // B: K=sA,   N=c0+lm
                bm.y = kvs[C_DIM + lm];         // B: K=sA+1, N=c0+lm
        __builtin_prefetch(kvs + 4 * C_DIM, 0, 1);  // global_prefetch_b8
        acc = __builtin_amdgcn_wmma_f32_16x16x4_f32(
            false, a, false, bm, (short)0, acc, false, false);
    }

    if (half == 0) {  // lanes 0-15 hold M=0..7 (real heads)
        for (int r = 0; r < 8; ++r)
            atomicAdd(&pkv[((size_t)bb * 16 + r) * C_DIM + c0 + lm], acc[r]);
    }
}

// ---------------------------------------------------------------------------
// Kernel 4: out[b,h,d] = pkv[b,h,:] @ Wv[:, h*64+d] + bv[h*64+d]
// (tiny: 16*8*64 outputs x 256 FMAs)
// ---------------------------------------------------------------------------
__global__ __launch_bounds__(64) void out_kernel(const float* __restrict__ pkv,
                                                 const float* __restrict__ Wv,
                                                 const float* __restrict__ bv,
                                                 float* __restrict__ out) {
    int b = blockIdx.x >> 3;
    int h = blockIdx.x & 7;
    int d = threadIdx.x;
    const float* pr   = pkv + ((size_t)b * 16 + h) * C_DIM;
    const float* wcol = Wv + h * 64 + d;
    float acc = bv[h * 64 + d];
    for (int c = 0; c < C_DIM; ++c)
        acc += pr[c] * wcol[(size_t)c * 512];
    out[((size_t)b * 8 + h) * 64 + d] = acc;
}

// ---------------------------------------------------------------------------
extern "C" void kernel_launch(void* const* d_in, const int* in_sizes, int n_in,
                              void* d_out, int out_size, void* d_ws, size_t ws_size,
                              hipStream_t stream) {
    const float* kv = (const float*)d_in[0];
    const float* Wk = (const float*)d_in[1];
    const float* bk = (const float*)d_in[2];
    const float* Wv = (const float*)d_in[3];
    const float* bv = (const float*)d_in[4];
    const float* q  = (const float*)d_in[5];
    float* out = (float*)d_out;

    // workspace layout (floats)
    float* w      = (float*)d_ws;
    float* wq     = w;            // 4096
    float* bq     = w + 4096;     // 16
    float* maxv   = w + 4112;     // 256
    float* denom  = w + 4368;     // 256
    float* pkv    = w + 4624;     // 65536
    float* scores = w + 70160;    // 2097152  (~8.7 MB total)

    hipMemsetAsync(pkv, 0, 65536 * sizeof(float), stream);

    prep_kernel<<<1, 256, 0, stream>>>(Wk, bk, q, wq, bq);
    scores_kernel<<<dim3(S_TOT / 128, B_DIM), 256, 0, stream>>>(kv, wq, bq, scores);
    stats_kernel<<<B_DIM * 8, 256, 0, stream>>>(scores, maxv, denom);
    pkv_kernel<<<dim3(C_DIM / 16, 8, B_DIM), 256, 0, stream>>>(kv, scores, maxv, denom, pkv);
    out_kernel<<<B_DIM * 8, 64, 0, stream>>>(pkv, Wv, bv, out);
}